// PCLoss_10179072491773
// MI455X (gfx1250) — compile-verified
//
#include <hip/hip_runtime.h>
#include <hip/hip_bf16.h>
#include <math.h>

typedef __attribute__((ext_vector_type(2))) float v2f;
typedef __attribute__((ext_vector_type(8))) float v8f;

#define BB 1024
#define DD 128
#define CC 1000

// ---------------------------------------------------------------------------
// Kernel A: logits[b, c] = x[b, :] . kernel[:, c] + bias[c]
// One wave computes a 16x64 tile (4 n-subtiles) with V_WMMA_F32_16X16X4_F32,
// looping K = 128 in steps of 4 and reusing the A fragment across the 4
// accumulators. Column indices are CLAMPED (not branched) so the inner loop
// is branch-free: EXEC stays all-1s, loads clause cleanly, no saveexec churn.
//
// ISA layouts (CDNA5 ISA, wave32):
//   A (16x4 f32):  lanes 0-15 -> M=lane, VGPR0=K0, VGPR1=K1
//                  lanes16-31 -> M=lane-16, VGPR0=K2, VGPR1=K3
//   B (4x16 f32):  mirror of A with N instead of M
//   C/D (16x16 f32, 8 VGPRs): VGPR v: lanes 0-15 -> M=v,  N=lane
//                                       lanes16-31 -> M=v+8, N=lane-16
// ---------------------------------------------------------------------------
__global__ __launch_bounds__(32) void logits_wmma_kernel(
    const float* __restrict__ x,      // [BB, DD]
    const float* __restrict__ w,      // [DD, CC]
    const float* __restrict__ bias,   // [CC]
    float* __restrict__ logits)       // [BB, CC] (workspace)
{
    const int lane = threadIdx.x;          // 0..31
    const int m0   = blockIdx.y * 16;      // batch-row tile base
    const int n0   = blockIdx.x * 64;      // class-col tile base (4 subtiles)
    const int half = lane >> 4;            // 0 -> K0,K1 ; 1 -> K2,K3
    const int l15  = lane & 15;
    const int row  = m0 + l15;             // A row this lane supplies

    const float* __restrict__ xrow = x + row * DD;

    int col[4], colc[4];
    #pragma unroll
    for (int j = 0; j < 4; ++j) {
        col[j]  = n0 + j * 16 + l15;
        colc[j] = (col[j] < CC) ? col[j] : (CC - 1);   // clamp: always in-bounds
    }

    v8f acc[4] = {{}, {}, {}, {}};

    #pragma unroll 4
    for (int k0 = 0; k0 < DD; k0 += 4) {
        const int k = k0 + half * 2;
        v2f a;
        a.x = xrow[k];
        a.y = xrow[k + 1];
        const float* __restrict__ wk0 = w + (size_t)k * CC;
        const float* __restrict__ wk1 = wk0 + CC;
        #pragma unroll
        for (int j = 0; j < 4; ++j) {
            v2f bf;
            bf.x = wk0[colc[j]];
            bf.y = wk1[colc[j]];
            // 8 args: (neg_a, A, neg_b, B, c_mod, C, reuse_a, reuse_b)
            acc[j] = __builtin_amdgcn_wmma_f32_16x16x4_f32(
                false, a, false, bf, (short)0, acc[j], false, false);
        }
    }

    const int rbase = m0 + half * 8;
    #pragma unroll
    for (int j = 0; j < 4; ++j) {
        if (col[j] < CC) {
            const float bv = bias[col[j]];
            #pragma unroll
            for (int v = 0; v < 8; ++v) {
                logits[(size_t)(rbase + v) * CC + col[j]] = acc[j][v] + bv;
            }
        }
    }
}

// ---------------------------------------------------------------------------
// Block-wide reductions over 256 threads (8 waves, wave32).
// ---------------------------------------------------------------------------
__device__ __forceinline__ float block_sum(float v, float* red, int t) {
    red[t] = v;
    __syncthreads();
    #pragma unroll
    for (int s = 128; s > 0; s >>= 1) {
        if (t < s) red[t] += red[t + s];
        __syncthreads();
    }
    float r = red[0];
    __syncthreads();
    return r;
}

__device__ __forceinline__ float block_max(float v, float* red, int t) {
    red[t] = v;
    __syncthreads();
    #pragma unroll
    for (int s = 128; s > 0; s >>= 1) {
        if (t < s) red[t] = fmaxf(red[t], red[t + s]);
        __syncthreads();
    }
    float r = red[0];
    __syncthreads();
    return r;
}

// ---------------------------------------------------------------------------
// Kernel B: one block per batch row b.
//   pcl1  = sum_d |x[b,d] - w_y[d]|
//   S_iw  = sum_c sum_d |x[b,d]  - kernel[d,c]|
//   S_ww  = sum_c sum_d |w_y[d]  - kernel[d,c]|
//   pcl2  = (S_iw + S_ww - pcl1) / (C-1)       (uses d_ww[b,y]=0, d_iw[b,y]=pcl1)
//   ce    = logsumexp(logits[b,:]) - logits[b,y]
//   out[b] = pcl1 - pcl2 + ce
//
// L1 sweep: blocked column mapping c = 4t..4t+3. 250 threads x 4 = exactly
// 1000 classes -> single uniform guard (t < 250), one 16B-aligned, fully
// coalesced global_load_b128 per thread per d instead of 4x b32.
// ---------------------------------------------------------------------------
__global__ __launch_bounds__(256) void pc_ce_kernel(
    const float* __restrict__ x,       // [BB, DD]
    const float* __restrict__ w,       // [DD, CC]
    const int*   __restrict__ label,   // [BB]
    const float* __restrict__ logits,  // [BB, CC]
    float* __restrict__ out)           // [BB]
{
    __shared__ float xs[DD];
    __shared__ float wys[DD];
    __shared__ float red[256];

    const int b = blockIdx.x;
    const int t = threadIdx.x;
    const int y = label[b];

    if (t < DD) {
        xs[t]  = x[b * DD + t];
        wys[t] = w[t * CC + y];   // w_y[d] = kernel[d, y]
    }
    __syncthreads();

    float siw_t = 0.0f;
    float sww_t = 0.0f;
    if (t < 250) {                       // threads 0..249 cover c = 4t..4t+3
        float acc_iw[4] = {0.f, 0.f, 0.f, 0.f};
        float acc_ww[4] = {0.f, 0.f, 0.f, 0.f};
        const float* __restrict__ kcol = w + 4 * t;   // byte offset 16t: aligned
        for (int d = 0; d < DD; ++d) {
            const float  xv = xs[d];
            const float  wv = wys[d];
            const float4 kv = *(const float4*)(kcol + (size_t)d * CC);
            acc_iw[0] += fabsf(xv - kv.x);
            acc_ww[0] += fabsf(wv - kv.x);
            acc_iw[1] += fabsf(xv - kv.y);
            acc_ww[1] += fabsf(wv - kv.y);
            acc_iw[2] += fabsf(xv - kv.z);
            acc_ww[2] += fabsf(wv - kv.z);
            acc_iw[3] += fabsf(xv - kv.w);
            acc_ww[3] += fabsf(wv - kv.w);
        }
        siw_t = (acc_iw[0] + acc_iw[1]) + (acc_iw[2] + acc_iw[3]);
        sww_t = (acc_ww[0] + acc_ww[1]) + (acc_ww[2] + acc_ww[3]);
    }

    const float S_iw = block_sum(siw_t, red, t);
    const float S_ww = block_sum(sww_t, red, t);

    // --- pcl1
    const float pv   = (t < DD) ? fabsf(xs[t] - wys[t]) : 0.0f;
    const float pcl1 = block_sum(pv, red, t);

    // --- cross-entropy from staged logits (stable log-softmax)
    const float* __restrict__ lrow = logits + (size_t)b * CC;
    float lmax_t = -INFINITY;
    for (int c = t; c < CC; c += 256) lmax_t = fmaxf(lmax_t, lrow[c]);
    const float lmax = block_max(lmax_t, red, t);

    float se_t = 0.0f;
    for (int c = t; c < CC; c += 256) se_t += expf(lrow[c] - lmax);
    const float se = block_sum(se_t, red, t);

    if (t == 0) {
        const float ly   = lrow[y];
        const float ce   = logf(se) + lmax - ly;
        const float pcl2 = (S_iw + S_ww - pcl1) / (float)(CC - 1);
        out[b] = pcl1 - pcl2 + ce;
    }
}

// ---------------------------------------------------------------------------
// Launch. Inputs (setup_inputs order): x [1024*128] f32, kernel [128*1000] f32,
// bias [1000] f32, label_idx [1024] i32. Output: [1024] f32.
// Workspace: logits stage, 1024*1000*4 = 4,096,000 bytes.
// ---------------------------------------------------------------------------
extern "C" void kernel_launch(void* const* d_in, const int* in_sizes, int n_in,
                              void* d_out, int out_size, void* d_ws, size_t ws_size,
                              hipStream_t stream) {
    const float* x      = (const float*)d_in[0];
    const float* w      = (const float*)d_in[1];
    const float* bias   = (const float*)d_in[2];
    const int*   label  = (const int*)d_in[3];
    float*       out    = (float*)d_out;
    float*       logits = (float*)d_ws;   // [BB, CC] f32 staging

    dim3 gridA((CC + 63) / 64, BB / 16);  // 16 x 64 tiles, 1 wave each
    logits_wmma_kernel<<<gridA, 32, 0, stream>>>(x, w, bias, logits);

    pc_ce_kernel<<<BB, 256, 0, stream>>>(x, w, label, logits, out);
}